// GCN_21260088115434
// MI455X (gfx1250) — compile-verified
//
#include <hip/hip_runtime.h>

#define NNODES 100000
#define NEDGES 1600000
#define NPAIRS 200000
#define DIN    128
#define DHID   128
#define DOUT   64
#define KDIM   128   // contraction dim is 128 for all three layers
#define APITCH 132   // padded LDS row pitch (floats): bank-conflict-free ds_load_b64

typedef __attribute__((ext_vector_type(2))) float v2f;
typedef __attribute__((ext_vector_type(8))) float v8f;

// ---------------- degree / normalization ----------------
__global__ void k_deg_init(float* __restrict__ deg, int n) {
  int i = blockIdx.x * blockDim.x + threadIdx.x;
  if (i < n) deg[i] = 1.0f;                       // self-loop contributes 1
}

__global__ void k_deg_accum(const int* __restrict__ dst, float* __restrict__ deg, int e) {
  int i = blockIdx.x * blockDim.x + threadIdx.x;
  if (i < e) atomicAdd(&deg[dst[i]], 1.0f);
}

__global__ void k_deg_rsqrt(float* __restrict__ deg, int n) {
  int i = blockIdx.x * blockDim.x + threadIdx.x;
  if (i < n) deg[i] = rsqrtf(deg[i]);             // deg >= 1 always
}

__global__ void k_edge_norm(const int* __restrict__ src, const int* __restrict__ dst,
                            const float* __restrict__ dinv, float* __restrict__ norm, int e) {
  int i = blockIdx.x * blockDim.x + threadIdx.x;
  if (i < e) norm[i] = dinv[src[i]] * dinv[dst[i]];
}

// ---------------- dense GEMM via V_WMMA_F32_16X16X4_F32 ----------------
// C[M x NOUT] = A[M x 128] @ B[128 x NOUT], NOUT compile-time (128 or 64) so
// every strided B access folds into the 24-bit immediate offset of
// global_load_b32 (max offset 127*128*4 = 65KB << 8MB range) -> no per-load
// 64-bit address arithmetic in the hot loop.
// Block = 128 threads (4 waves). Block owns one 16-row M tile; the 16x128 A
// tile is a CONTIGUOUS 8KB region of A -> coalesced cooperative copy to LDS
// (padded pitch 132 floats => conflict-free ds_load_2addr_b64 in the MAC loop),
// shared by all 4 waves. Each wave computes one 16x16 C tile (4 N-tiles/block).
// A 16x4 lane layout: lane<16 -> K={k,k+1}, lane>=16 -> K={k+2,k+3}, row=lane&15.
// B 4x16 lane layout: v0 = rows {k,k+2}, v1 = rows {k+1,k+3}, col = lane&15.
// C 16x16 layout: VGPR r -> row r + 8*(lane>=16), col = lane&15.
template <int NOUT>
__global__ __launch_bounds__(128) void k_gemm_wmma(const float* __restrict__ A,
                                                   const float* __restrict__ B,
                                                   float* __restrict__ C) {
  __shared__ float lA[16 * APITCH];                // 8.25 KB

  const int tid  = threadIdx.x;                    // 0..127
  const int wave = tid >> 5;
  const int lane = tid & 31;
  const int row0 = blockIdx.x * 16;
  const int col0 = (blockIdx.y * 4 + wave) * 16;
  const int half = lane >> 4;                      // 0: lanes 0-15, 1: lanes 16-31
  const int l16  = lane & 15;

  // --- stage A tile: 2048 floats = 512 float4, 4 per thread, fully coalesced ---
  {
    const float4* gA = (const float4*)(A + (size_t)row0 * KDIM);  // contiguous 8KB
    float4* sA = (float4*)lA;                                     // APITCH/4 = 33
#pragma unroll
    for (int i = 0; i < 4; ++i) {
      int idx = tid + i * 128;                     // float4 index 0..511
      int r   = idx >> 5;                          // row (32 float4 per row)
      int c4  = idx & 31;
      sA[r * (APITCH / 4) + c4] = gA[idx];
    }
  }
  __syncthreads();

  const float* aP = lA + l16 * APITCH + half * 2;
  const float* bP = B + (half * 2) * NOUT + (col0 + l16);  // loop-invariant base

  v8f acc = {};
#pragma unroll
  for (int k = 0; k < KDIM; k += 4) {
    v2f av = *(const v2f*)(aP + k);                // ds_load, conflict-free
    v2f bv;
    bv.x = bP[k * NOUT];                           // immediate-offset global loads
    bv.y = bP[(k + 1) * NOUT];
    // 8 args: (neg_a, A, neg_b, B, c_mod, C, reuse_a, reuse_b)
    acc = __builtin_amdgcn_wmma_f32_16x16x4_f32(false, av, false, bv,
                                                (short)0, acc, false, false);
  }

  float* cP = C + (size_t)(row0 + half * 8) * NOUT + (col0 + l16);
#pragma unroll
  for (int r = 0; r < 8; ++r) cP[r * NOUT] = acc[r];       // immediate offsets
}

// ---------------- aggregation: out = selfloop*h + bias, then scatter edges ----
__global__ void k_agg_init(const float* __restrict__ h, const float* __restrict__ dinv,
                           const float* __restrict__ bias, float* __restrict__ out,
                           int n, int fshift, int fmask) {
  long long idx = (long long)blockIdx.x * blockDim.x + threadIdx.x;
  if (idx >= ((long long)n << fshift)) return;
  int i = (int)(idx >> fshift);
  int f = (int)(idx & fmask);
  float s = dinv[i];
  out[idx] = s * s * h[idx] + bias[f];
}

// one float4 of features per thread; 4 global fp32 atomics to scatter
__global__ void k_agg_edges(const float* __restrict__ h, const int* __restrict__ src,
                            const int* __restrict__ dst, const float* __restrict__ norm,
                            float* __restrict__ out, int e, int f4shift, int f4mask) {
  long long idx = (long long)blockIdx.x * blockDim.x + threadIdx.x;
  if (idx >= ((long long)e << f4shift)) return;
  int ei = (int)(idx >> f4shift);
  int f4 = (int)(idx & f4mask);
  int s = src[ei], d = dst[ei];
  float w = norm[ei];
  const float4 v = ((const float4*)h)[((size_t)s << f4shift) + f4];
  float* o = out + ((((size_t)d << f4shift) + (size_t)f4) << 2);
  atomicAdd(o + 0, v.x * w);
  atomicAdd(o + 1, v.y * w);
  atomicAdd(o + 2, v.z * w);
  atomicAdd(o + 3, v.w * w);
}

__global__ void k_relu(float* __restrict__ x, long long n) {
  long long i = (long long)blockIdx.x * blockDim.x + threadIdx.x;
  if (i < n) x[i] = fmaxf(x[i], 0.0f);
}

// ---------------- decoder: out[p] = dot(z[a_p], z[b_p]) over 64 dims --------
// 16 lanes per pair, float4 loads, shfl_xor tree reduce within the 16-group.
__global__ void k_decode(const float* __restrict__ z, const int* __restrict__ pa,
                         const int* __restrict__ pb, float* __restrict__ out, int np) {
  int tid = blockIdx.x * blockDim.x + threadIdx.x;
  int pair = tid >> 4;
  int l    = tid & 15;
  if (pair >= np) return;
  const float4 va = ((const float4*)z)[((size_t)pa[pair] << 4) + l];
  const float4 vb = ((const float4*)z)[((size_t)pb[pair] << 4) + l];
  float p = va.x * vb.x + va.y * vb.y + va.z * vb.z + va.w * vb.w;
  p += __shfl_xor(p, 8);
  p += __shfl_xor(p, 4);
  p += __shfl_xor(p, 2);
  p += __shfl_xor(p, 1);
  if (l == 0) out[pair] = p;
}

// ---------------- host side ----------------
extern "C" void kernel_launch(void* const* d_in, const int* in_sizes, int n_in,
                              void* d_out, int out_size, void* d_ws, size_t ws_size,
                              hipStream_t stream) {
  (void)in_sizes; (void)n_in; (void)out_size; (void)ws_size;

  const float* x  = (const float*)d_in[0];
  const int*   ei = (const int*)d_in[1];     // [2, E] row-major
  const int*   el = (const int*)d_in[2];     // [2, EL]
  const float* W1 = (const float*)d_in[3];
  const float* b1 = (const float*)d_in[4];
  const float* W2 = (const float*)d_in[5];
  const float* b2 = (const float*)d_in[6];
  const float* W3 = (const float*)d_in[7];
  const float* b3 = (const float*)d_in[8];

  const int* src = ei;
  const int* dst = ei + NEDGES;
  const int* pa  = el;
  const int* pb  = el + NPAIRS;
  float* out = (float*)d_out;

  // workspace carve-out (fp32): dinv[N] | norm[E] | bufA[N*128] | bufB[N*128]
  float* dinv = (float*)d_ws;
  float* norm = dinv + ((NNODES + 127) & ~127);
  float* bufA = norm + ((NEDGES + 127) & ~127);
  float* bufB = bufA + (size_t)NNODES * DHID;

  auto cdiv = [](long long a, long long b) { return (int)((a + b - 1) / b); };

  // ---- gcn_norm ----
  k_deg_init <<<cdiv(NNODES, 256), 256, 0, stream>>>(dinv, NNODES);
  k_deg_accum<<<cdiv(NEDGES, 256), 256, 0, stream>>>(dst, dinv, NEDGES);
  k_deg_rsqrt<<<cdiv(NNODES, 256), 256, 0, stream>>>(dinv, NNODES);
  k_edge_norm<<<cdiv(NEDGES, 256), 256, 0, stream>>>(src, dst, dinv, norm, NEDGES);

  const int mTiles = NNODES / 16;         // 6250, exact

  // ---- layer 1: h = relu(agg(x @ W1) + b1) ----
  k_gemm_wmma<DHID><<<dim3(mTiles, DHID / 64), 128, 0, stream>>>(x, W1, bufA);
  k_agg_init <<<cdiv((long long)NNODES * DHID, 256), 256, 0, stream>>>(bufA, dinv, b1, bufB, NNODES, 7, 127);
  k_agg_edges<<<cdiv((long long)NEDGES * (DHID / 4), 256), 256, 0, stream>>>(bufA, src, dst, norm, bufB, NEDGES, 5, 31);
  k_relu     <<<cdiv((long long)NNODES * DHID, 256), 256, 0, stream>>>(bufB, (long long)NNODES * DHID);

  // ---- layer 2 ----
  k_gemm_wmma<DHID><<<dim3(mTiles, DHID / 64), 128, 0, stream>>>(bufB, W2, bufA);
  k_agg_init <<<cdiv((long long)NNODES * DHID, 256), 256, 0, stream>>>(bufA, dinv, b2, bufB, NNODES, 7, 127);
  k_agg_edges<<<cdiv((long long)NEDGES * (DHID / 4), 256), 256, 0, stream>>>(bufA, src, dst, norm, bufB, NEDGES, 5, 31);
  k_relu     <<<cdiv((long long)NNODES * DHID, 256), 256, 0, stream>>>(bufB, (long long)NNODES * DHID);

  // ---- layer 3 (no relu): z = agg(h @ W3) + b3 ----
  k_gemm_wmma<DOUT><<<dim3(mTiles, DOUT / 64), 128, 0, stream>>>(bufB, W3, bufA);
  k_agg_init <<<cdiv((long long)NNODES * DOUT, 256), 256, 0, stream>>>(bufA, dinv, b3, bufB, NNODES, 6, 63);
  k_agg_edges<<<cdiv((long long)NEDGES * (DOUT / 4), 256), 256, 0, stream>>>(bufA, src, dst, norm, bufB, NEDGES, 4, 15);

  // ---- decode ----
  k_decode<<<cdiv((long long)NPAIRS * 16, 256), 256, 0, stream>>>(bufB, pa, pb, out, NPAIRS);
}